// ChunkLayerReference_38422777430600
// MI455X (gfx1250) — compile-verified
//
#include <hip/hip_runtime.h>
#include <hip/hip_bf16.h>

#ifndef __has_builtin
#define __has_builtin(x) 0
#endif

#if defined(__AMDGCN__) &&                                         \
    __has_builtin(__builtin_amdgcn_global_load_async_to_lds_b128) && \
    __has_builtin(__builtin_amdgcn_global_store_async_from_lds_b128) && \
    __has_builtin(__builtin_amdgcn_s_wait_asynccnt)
#define CHUNK_USE_ASYNC 1
#else
#define CHUNK_USE_ASYNC 0
#endif

#if CHUNK_USE_ASYNC
// The gfx1250 async builtins take pointers to a 16-byte int vector in
// global (AS1) and LDS (AS3) address spaces (per hipcc diagnostic).
typedef int v4i __attribute__((vector_size(16)));
typedef __attribute__((address_space(1))) v4i* g_v4i_p;
typedef __attribute__((address_space(3))) v4i* l_v4i_p;

__device__ __forceinline__ g_v4i_p to_global_v4(const void* p) {
  // AS1 pointer has the same 64-bit value as the generic pointer.
  return (g_v4i_p)(unsigned long long)p;
}
__device__ __forceinline__ l_v4i_p to_lds_v4(void* p) {
  // generic LDS pointer = {shared_base[63:32], lds_offset[31:0]}; AS3 = low 32 bits.
  return (l_v4i_p)(unsigned int)(unsigned long long)p;
}
#endif

#define SCAN_THREADS 256

// ---------------------------------------------------------------------------
// Kernel 1: per-batch stable partition of indices (boundary tokens first,
// then non-boundary tokens, both in original order), truncated to M entries.
// Also writes the float mask output: mask_out[b, j] = (j < num_tokens[b]).
// One block (256 threads) per batch; block-wide prefix sum in LDS.
// ---------------------------------------------------------------------------
__global__ void __launch_bounds__(SCAN_THREADS)
build_order_kernel(const int* __restrict__ mask, int* __restrict__ order,
                   float* __restrict__ mask_out, int L, int M) {
  __shared__ int s_cnt[SCAN_THREADS];
  const int b = blockIdx.x;
  const int t = threadIdx.x;
  const int chunk = L / SCAN_THREADS;            // 8192 / 256 = 32
  const int start = t * chunk;
  const int* m = mask + (size_t)b * L;

  // Pass 1: local count of boundary tokens in this thread's chunk
  int cnt = 0;
  for (int k = 0; k < chunk; ++k)
    cnt += (m[start + k] != 0) ? 1 : 0;
  s_cnt[t] = cnt;
  __syncthreads();

  // Hillis-Steele inclusive scan over the 256 per-thread counts
  for (int off = 1; off < SCAN_THREADS; off <<= 1) {
    int add = (t >= off) ? s_cnt[t - off] : 0;
    __syncthreads();
    s_cnt[t] += add;
    __syncthreads();
  }
  const int incl  = s_cnt[t];
  const int excl  = incl - cnt;                  // boundary tokens before this chunk
  const int total = s_cnt[SCAN_THREADS - 1];     // num_tokens[b]

  // Pass 2: emit stable-partition positions
  int bpos  = excl;                              // next boundary slot
  int nbpos = start - excl;                      // non-boundary tokens before this chunk
  int* ord = order + (size_t)b * M;
  for (int k = 0; k < chunk; ++k) {
    const int i = start + k;
    if (m[i] != 0) {
      if (bpos < M) ord[bpos] = i;
      ++bpos;
    } else {
      const int p = total + nbpos;
      if (p < M) ord[p] = i;
      ++nbpos;
    }
  }

  // Mask output (float: 1.0 valid, 0.0 padding)
  float* mo = mask_out + (size_t)b * M;
  for (int j = t; j < M; j += SCAN_THREADS)
    mo[j] = (j < total) ? 1.0f : 0.0f;
}

// ---------------------------------------------------------------------------
// Kernel 2: gather rows. One block per output row (4 KB), 256 threads, each
// thread moves one float4. On CDNA5 this uses the async global<->LDS DMA path
// (GLOBAL_LOAD_ASYNC_TO_LDS_B128 / GLOBAL_STORE_ASYNC_FROM_LDS_B128,
// ASYNCcnt-tracked) so row data never transits the VGPR file.
// ---------------------------------------------------------------------------
__global__ void __launch_bounds__(256)
gather_rows_kernel(const float* __restrict__ src, const int* __restrict__ order,
                   float* __restrict__ dst, int M, int total, int L, int D) {
  const int t = threadIdx.x;
#if CHUNK_USE_ASYNC
  __shared__ __align__(16) float lbuf[256 * 4];  // 4 KB staging tile
#endif
  for (int row = blockIdx.x; row < total; row += gridDim.x) {
    const int b = row / M;
    const int s = order[row];                    // source sequence position
    const float* sp = src + ((size_t)b * L + (size_t)s) * D;
    float*       dp = dst + (size_t)row * D;
    for (int c = t * 4; c < D; c += 256 * 4) {
#if CHUNK_USE_ASYNC
      __builtin_amdgcn_global_load_async_to_lds_b128(
          to_global_v4(sp + c), to_lds_v4(&lbuf[t * 4]), 0, 0);
      __builtin_amdgcn_s_wait_asynccnt(0);       // LDS tile valid for this wave
      __builtin_amdgcn_global_store_async_from_lds_b128(
          to_global_v4(dp + c), to_lds_v4(&lbuf[t * 4]), 0, 0);
      __builtin_amdgcn_s_wait_asynccnt(0);       // safe to reuse LDS tile
#else
      float4 v = *reinterpret_cast<const float4*>(sp + c);
      *reinterpret_cast<float4*>(dp + c) = v;
#endif
    }
  }
}

extern "C" void kernel_launch(void* const* d_in, const int* in_sizes, int n_in,
                              void* d_out, int out_size, void* d_ws, size_t ws_size,
                              hipStream_t stream) {
  (void)n_in; (void)ws_size;
  const float* hidden = (const float*)d_in[0];
  const int*   bmask  = (const int*)d_in[1];

  const int B  = 8;                        // per reference setup
  const int BL = in_sizes[1];              // B * L
  const int L  = BL / B;                   // 8192
  const int D  = in_sizes[0] / BL;         // 1024
  const int M  = out_size / (B * (D + 1)); // out = B*M*D hidden + B*M mask

  if (M <= 0) return;

  float* out_hidden = (float*)d_out;
  float* out_mask   = (float*)d_out + (size_t)B * M * D;
  int*   order      = (int*)d_ws;          // B*M ints of scratch

  build_order_kernel<<<B, SCAN_THREADS, 0, stream>>>(bmask, order, out_mask, L, M);

  const int total = B * M;                 // number of gathered rows
  int blocks = total;                      // one 4KB row per block
  gather_rows_kernel<<<blocks, 256, 0, stream>>>(hidden, order, out_hidden, M, total, L, D);
}